// SimpleGATLayer_70007966925386
// MI455X (gfx1250) — compile-verified
//
#include <hip/hip_runtime.h>

#define N_NODES   50000
#define N_EDGES   1600000
#define IN_F      128
#define HEADS     4
#define HEAD_DIM  32
#define OUT_F     (HEADS * HEAD_DIM)   // 128
#define NEG_SLOPE 0.2f

typedef float v2f __attribute__((ext_vector_type(2)));
typedef float v8f __attribute__((ext_vector_type(8)));

// ---------------------------------------------------------------------------
// Kernel 1: h = x @ W^T via V_WMMA_F32_16X16X4_F32.
// One wave computes one 16x16 tile of h, accumulating K=128 in 32 WMMA steps.
// Fragment layouts (CDNA5 ISA 7.12.2, fp32):
//   A 16x4 : lane L, elem v -> row m0+(L&15),            K = kk + 2*(L>>4) + v
//   B 4x16 : lane L, elem v -> col n0+(L&15) (=W row),   K = kk + 2*(L>>4) + v
//   C/D    : lane L, elem v -> M = m0 + v + 8*(L>>4), N = n0 + (L&15)
// ---------------------------------------------------------------------------
__global__ void gat_gemm_wmma(const float* __restrict__ x,
                              const float* __restrict__ W,
                              float* __restrict__ h) {
  const int wave = (blockIdx.x * blockDim.x + threadIdx.x) >> 5;
  const int lane = threadIdx.x & 31;
  const int ROW_TILES = N_NODES / 16;   // 3125 (50000 divisible by 16)
  const int COL_TILES = OUT_F / 16;     // 8
  if (wave >= ROW_TILES * COL_TILES) return;

  const int tr   = wave / COL_TILES;
  const int tc   = wave % COL_TILES;
  const int m0   = tr * 16;
  const int n0   = tc * 16;
  const int half = lane >> 4;           // 0: lanes 0-15, 1: lanes 16-31
  const int l15  = lane & 15;
  const int koff = half * 2;

  const float* __restrict__ xrow = x + (size_t)(m0 + l15) * IN_F + koff;
  const float* __restrict__ wrow = W + (size_t)(n0 + l15) * IN_F + koff;

  v8f acc = {};
#pragma unroll
  for (int kk = 0; kk < IN_F; kk += 4) {
    v2f a;
    a.x = xrow[kk];
    a.y = xrow[kk + 1];
    v2f b;
    b.x = wrow[kk];
    b.y = wrow[kk + 1];
    // (neg_a, A, neg_b, B, c_mod, C, reuse_a, reuse_b)
    acc = __builtin_amdgcn_wmma_f32_16x16x4_f32(false, a, false, b,
                                                (short)0, acc, false, false);
  }

#pragma unroll
  for (int v = 0; v < 8; ++v) {
    h[(size_t)(m0 + v + 8 * half) * OUT_F + n0 + l15] = acc[v];
  }
}

// ---------------------------------------------------------------------------
// Kernel 2: per-node attention scalars
//   alpha_src[n*H+h] = dot(h[n,h,:], a[h, 0:32])
//   alpha_dst[n*H+h] = dot(h[n,h,:], a[h,32:64])
// One thread per (node, head); float4 vectorized dot of length 32.
// ---------------------------------------------------------------------------
__global__ void gat_alpha(const float* __restrict__ h,
                          const float* __restrict__ a,
                          float* __restrict__ alpha_src,
                          float* __restrict__ alpha_dst) {
  const int t = blockIdx.x * blockDim.x + threadIdx.x;
  if (t >= N_NODES * HEADS) return;
  const int head = t & (HEADS - 1);
  const int node = t >> 2;

  const float4* __restrict__ hv = (const float4*)(h + (size_t)node * OUT_F + head * HEAD_DIM);
  const float4* __restrict__ as = (const float4*)(a + head * 2 * HEAD_DIM);
  const float4* __restrict__ ad = (const float4*)(a + head * 2 * HEAD_DIM + HEAD_DIM);

  float s = 0.f, d = 0.f;
#pragma unroll
  for (int j = 0; j < HEAD_DIM / 4; ++j) {
    const float4 hj = hv[j];
    const float4 sj = as[j];
    const float4 dj = ad[j];
    s += hj.x * sj.x + hj.y * sj.y + hj.z * sj.z + hj.w * sj.w;
    d += hj.x * dj.x + hj.y * dj.y + hj.z * dj.z + hj.w * dj.w;
  }
  alpha_src[t] = s;
  alpha_dst[t] = d;
}

// ---------------------------------------------------------------------------
// Kernel 3: zero the output accumulator (harness poisons d_out).
// ---------------------------------------------------------------------------
__global__ void gat_zero(float4* __restrict__ out, int n4) {
  const int t = blockIdx.x * blockDim.x + threadIdx.x;
  if (t < n4) out[t] = make_float4(0.f, 0.f, 0.f, 0.f);
}

// ---------------------------------------------------------------------------
// Kernel 4: edge scatter. One wave per edge; lane l owns output dims
// [4l, 4l+3] (head = l>>3). att = leakyrelu(alpha_src[src,h]+alpha_dst[dst,h]);
// out[dst] += att * h[src] via hardware global_atomic_add_f32.
// h / alpha / out are all L2-resident (< 53 MB total vs 192 MB L2).
// ---------------------------------------------------------------------------
__global__ void gat_edge(const long long* __restrict__ ei,
                         const float* __restrict__ h,
                         const float* __restrict__ alpha_src,
                         const float* __restrict__ alpha_dst,
                         float* __restrict__ out) {
  const int wave = (blockIdx.x * blockDim.x + threadIdx.x) >> 5;
  const int lane = threadIdx.x & 31;
  if (wave >= N_EDGES) return;

  const int src  = (int)ei[wave];
  const int dst  = (int)ei[N_EDGES + wave];
  const int head = lane >> 3;

  float att = alpha_src[src * HEADS + head] + alpha_dst[dst * HEADS + head];
  att = (att > 0.f) ? att : NEG_SLOPE * att;

  const float4 hs = *(const float4*)(h + (size_t)src * OUT_F + lane * 4);
  float* __restrict__ ob = out + (size_t)dst * OUT_F + lane * 4;

  unsafeAtomicAdd(ob + 0, att * hs.x);
  unsafeAtomicAdd(ob + 1, att * hs.y);
  unsafeAtomicAdd(ob + 2, att * hs.z);
  unsafeAtomicAdd(ob + 3, att * hs.w);
}

// ---------------------------------------------------------------------------
extern "C" void kernel_launch(void* const* d_in, const int* in_sizes, int n_in,
                              void* d_out, int out_size, void* d_ws, size_t ws_size,
                              hipStream_t stream) {
  const float*     x  = (const float*)d_in[0];
  const long long* ei = (const long long*)d_in[1];   // int64 edge_index (2, E)
  const float*     W  = (const float*)d_in[2];
  const float*     a  = (const float*)d_in[3];
  float*           out = (float*)d_out;

  // Workspace layout (fp32): h [N*128] | alpha_src [N*4] | alpha_dst [N*4]
  float* h         = (float*)d_ws;
  float* alpha_src = h + (size_t)N_NODES * OUT_F;
  float* alpha_dst = alpha_src + (size_t)N_NODES * HEADS;

  // 1) h = x @ W^T : 25000 waves, 8 waves (256 thr) per block.
  gat_gemm_wmma<<<(N_NODES / 16) * (OUT_F / 16) / 8, 256, 0, stream>>>(x, W, h);

  // 2) node-level attention scalars.
  gat_alpha<<<(N_NODES * HEADS + 255) / 256, 256, 0, stream>>>(h, a, alpha_src, alpha_dst);

  // 3) zero output.
  const int n4 = N_NODES * OUT_F / 4;
  gat_zero<<<(n4 + 255) / 256, 256, 0, stream>>>((float4*)out, n4);

  // 4) edge scatter: one wave per edge -> E/8 blocks of 256 threads.
  gat_edge<<<N_EDGES / 8, 256, 0, stream>>>(ei, h, alpha_src, alpha_dst, out);
}